// GCN_73005854098237
// MI455X (gfx1250) — compile-verified
//
#include <hip/hip_runtime.h>

typedef __attribute__((ext_vector_type(2))) float v2f;
typedef __attribute__((ext_vector_type(8))) float v8f;

#define N_NODES   100000
#define N_EDGES   1000000
#define NUM_GRAPHS 128
#define HID        128

// ---------------- degree computation ----------------
__global__ void gcn_deg_init(float* __restrict__ deg) {
  int i = blockIdx.x * 256 + threadIdx.x;
  if (i < N_NODES) deg[i] = 1.0f;                 // self-loop contributes 1
}

__global__ void gcn_deg_accum(const int* __restrict__ dst, float* __restrict__ deg) {
  int i = blockIdx.x * 256 + threadIdx.x;
  if (i < N_EDGES) atomicAdd(&deg[dst[i]], 1.0f);
}

__global__ void gcn_deg_rsqrt(float* __restrict__ deg) {
  int i = blockIdx.x * 256 + threadIdx.x;
  if (i < N_NODES) deg[i] = rsqrtf(deg[i]);       // deg >= 1 always (self-loops)
}

// ---------------- fp32 WMMA GEMM: H[N,128] = X[N,K] @ W[K,128] ----------------
// One wave per 16x16 tile; 8 waves/block cover all 128 output columns of a
// 16-row stripe. V_WMMA_F32_16X16X4_F32, K-loop step 4.
__global__ void gcn_gemm_wmma(const float* __restrict__ X, const float* __restrict__ W,
                              float* __restrict__ H, int K) {
  const int lane = threadIdx.x & 31;
  const int wv   = threadIdx.x >> 5;   // 0..7 -> N-tile
  const int m0   = blockIdx.x << 4;
  const int n0   = wv << 4;
  const int hi   = lane >> 4;          // lane-half selects K pair
  const int lo   = lane & 15;

  const float* __restrict__ xrow = X + (long)(m0 + lo) * K;
  v8f acc = {};
  for (int k0 = 0; k0 < K; k0 += 4) {
    const int ka = k0 + (hi << 1);
    v2f a, b;
    a.x = xrow[ka];                               // A: 16x4, K = 2*hi + vgpr
    a.y = xrow[ka + 1];
    b.x = W[(long)ka * HID + n0 + lo];            // B: 4x16, same K split, N = lo
    b.y = W[(long)(ka + 1) * HID + n0 + lo];
    acc = __builtin_amdgcn_wmma_f32_16x16x4_f32(
        /*neg_a=*/false, a, /*neg_b=*/false, b,
        /*c_mod=*/(short)0, acc, /*reuse_a=*/false, /*reuse_b=*/false);
  }
  // D layout: VGPR r -> row m0 + r + 8*hi, col n0 + lo
  float* __restrict__ hcol = H + (long)m0 * HID + n0 + lo;
#pragma unroll
  for (int r = 0; r < 8; ++r)
    hcol[(long)(r + (hi << 3)) * HID] = acc[r];
}

// ---------------- helpers ----------------
__global__ void gcn_zero(float* __restrict__ p, long n) {
  long i = (long)blockIdx.x * 256 + threadIdx.x;
  if (i < n) p[i] = 0.0f;
}

// one wave per edge: float4 gather of h[src], 4 f32 atomic adds into out[dst]
__global__ void gcn_scatter(const int* __restrict__ src, const int* __restrict__ dst,
                            const float* __restrict__ dis, const float* __restrict__ h,
                            float* __restrict__ out) {
  const int e = blockIdx.x * 8 + (threadIdx.x >> 5);
  if (e >= N_EDGES) return;
  const int lane = threadIdx.x & 31;
  const int s = src[e], d = dst[e];
  const float nrm = dis[s] * dis[d];
  const float4 v = ((const float4*)(h + (long)s * HID))[lane];
  float* o = out + (long)d * HID + lane * 4;
  atomicAdd(o + 0, nrm * v.x);
  atomicAdd(o + 1, nrm * v.y);
  atomicAdd(o + 2, nrm * v.z);
  atomicAdd(o + 3, nrm * v.w);
}

// agg = relu(agg + d^-1 * h (self-loop) + bias)
__global__ void gcn_finalize_relu(float* __restrict__ agg, const float* __restrict__ h,
                                  const float* __restrict__ dis, const float* __restrict__ bias) {
  long i = (long)blockIdx.x * 256 + threadIdx.x;
  if (i >= (long)N_NODES * HID) return;
  const int node = (int)(i >> 7);
  const int c    = (int)(i & (HID - 1));
  const float dii = dis[node];
  const float v = agg[i] + dii * dii * h[i] + bias[c];
  agg[i] = fmaxf(v, 0.0f);
}

// per-graph sum + count (batch is sorted, but atomics are simplest & cheap here)
__global__ void gcn_pool(const float* __restrict__ h, const int* __restrict__ batch,
                         float* __restrict__ sums, float* __restrict__ cnt) {
  const int node = blockIdx.x * 2 + (threadIdx.x >> 7);
  if (node >= N_NODES) return;
  const int c = threadIdx.x & 127;
  const int g = batch[node];
  atomicAdd(&sums[(long)g * HID + c], h[(long)node * HID + c]);
  if (c == 0) atomicAdd(&cnt[g], 1.0f);
}

__global__ void gcn_classifier(const float* __restrict__ sums, const float* __restrict__ cnt,
                               const float* __restrict__ Wc, const float* __restrict__ bc,
                               float* __restrict__ out) {
  const int g = blockIdx.x;
  const int o = threadIdx.x;
  if (o >= 10) return;
  const float inv = 1.0f / fmaxf(cnt[g], 1.0f);
  float acc = 0.0f;
  for (int k = 0; k < HID; ++k)
    acc = fmaf(sums[(long)g * HID + k], Wc[k * 10 + o], acc);
  out[g * 10 + o] = acc * inv + bc[o];
}

// ---------------- host orchestration ----------------
extern "C" void kernel_launch(void* const* d_in, const int* in_sizes, int n_in,
                              void* d_out, int out_size, void* d_ws, size_t ws_size,
                              hipStream_t stream) {
  const float* x     = (const float*)d_in[0];
  const int*   ei    = (const int*)d_in[1];     // [2, E] flattened
  const int*   batch = (const int*)d_in[2];
  const float* W1 = (const float*)d_in[3];
  const float* b1 = (const float*)d_in[4];
  const float* W2 = (const float*)d_in[5];
  const float* b2 = (const float*)d_in[6];
  const float* W3 = (const float*)d_in[7];
  const float* b3 = (const float*)d_in[8];
  const float* Wc = (const float*)d_in[9];
  const float* bc = (const float*)d_in[10];
  float* out = (float*)d_out;

  const long NH = (long)N_NODES * HID;          // 12.8M floats
  float* ws   = (float*)d_ws;
  float* deg  = ws;                             // [N] -> deg_isqrt in place
  float* bufA = ws + N_NODES;                   // [N,128] GEMM output (reused)
  float* bufB = bufA + NH;                      // [N,128] aggregation / layer out
  float* sums = bufB + NH;                      // [G,128]
  float* cnt  = sums + (long)NUM_GRAPHS * HID;  // [G]

  const int* esrc = ei;
  const int* edst = ei + N_EDGES;

  const dim3 b256(256);

  // normalized degrees (shared by all layers)
  gcn_deg_init <<<(N_NODES + 255) / 256, b256, 0, stream>>>(deg);
  gcn_deg_accum<<<(N_EDGES + 255) / 256, b256, 0, stream>>>(edst, deg);
  gcn_deg_rsqrt<<<(N_NODES + 255) / 256, b256, 0, stream>>>(deg);

  const float* layer_in = x;
  const float* Ws[3] = {W1, W2, W3};
  const float* bs[3] = {b1, b2, b3};
  const int    Ks[3] = {64, 128, 128};

  for (int l = 0; l < 3; ++l) {
    gcn_gemm_wmma<<<N_NODES / 16, b256, 0, stream>>>(layer_in, Ws[l], bufA, Ks[l]);
    gcn_zero<<<(int)((NH + 255) / 256), b256, 0, stream>>>(bufB, NH);
    gcn_scatter<<<(N_EDGES + 7) / 8, b256, 0, stream>>>(esrc, edst, deg, bufA, bufB);
    gcn_finalize_relu<<<(int)((NH + 255) / 256), b256, 0, stream>>>(bufB, bufA, deg, bs[l]);
    layer_in = bufB;                            // aggregation buffer feeds next GEMM
  }

  gcn_zero<<<(NUM_GRAPHS * HID + NUM_GRAPHS + 255) / 256, b256, 0, stream>>>(
      sums, (long)NUM_GRAPHS * HID + NUM_GRAPHS);
  gcn_pool<<<(N_NODES + 1) / 2, b256, 0, stream>>>(bufB, batch, sums, cnt);
  gcn_classifier<<<NUM_GRAPHS, dim3(32), 0, stream>>>(sums, cnt, Wc, bc, out);
}